// Seq2SeqRNNWithAttention_11115375362491
// MI455X (gfx1250) — compile-verified
//
#include <hip/hip_runtime.h>
#include <cstddef>

// ---------------------------------------------------------------------------
// Seq2Seq RNN + Bahdanau attention for MI455X (gfx1250, wave32, WMMA).
// All GEMMs use V_WMMA_F32_16X16X4_F32 (exact fp32 to match the reference).
// ---------------------------------------------------------------------------

typedef float v2f __attribute__((ext_vector_type(2)));
typedef float v8f __attribute__((ext_vector_type(8)));

#define BATCH 32
#define SRCLEN 128
#define TGTLEN 64
#define HID 512
#define EMB 512
#define VOCAB 32000

__device__ __forceinline__ v8f wmma4(v2f a, v2f b, v8f c) {
  // D = A(16x4 f32) * B(4x16 f32) + C(16x16 f32)
  return __builtin_amdgcn_wmma_f32_16x16x4_f32(false, a, false, b, (short)0, c,
                                               false, false);
}

__device__ __forceinline__ v8f v8zero() {
  v8f z = {0.f, 0.f, 0.f, 0.f, 0.f, 0.f, 0.f, 0.f};
  return z;
}

// ---------------------------------------------------------------------------
// Generic fp32 WMMA GEMM:  C[r,n] = (gather?)A[r,:] @ B[:,n] + bias1 + bias2
// Block = 256 threads (8 waves).  Macro tile: 32 (M) x 128 (N).
// Each wave owns one 16-row strip and two 16-col tiles (2 x v8f accumulators).
// R % 32 == 0, N % 128 == 0, K % 4 == 0 for every call site.
// ---------------------------------------------------------------------------
__global__ __launch_bounds__(256) void gemm_tile(
    const float* __restrict__ A, const int* __restrict__ idx, int strideA,
    const float* __restrict__ Bm, int ldb, const float* __restrict__ bias1,
    const float* __restrict__ bias2, float* __restrict__ C, int ldc, int R,
    int K, int N) {
  const int lane = threadIdx.x & 31;
  const int wave = threadIdx.x >> 5;  // 0..7
  const int mt = wave >> 2;           // 0..1  (which 16-row strip)
  const int ntb = (wave & 3) * 2;     // 0,2,4,6 (pair of 16-col tiles)
  const int m0 = blockIdx.x * 32;
  const int n0 = blockIdx.y * 128;
  const int l15 = lane & 15;
  const int koff = (lane >> 4) * 2;  // 0 or 2 (K sub-offset per half-wave)

  const int am = m0 + mt * 16 + l15;  // logical A row for this lane
  const long arow = idx ? (long)idx[am] : (long)am;
  const float* __restrict__ Arow = A + arow * (long)strideA;

  const int bn0 = n0 + (ntb + 0) * 16 + l15;
  const int bn1 = n0 + (ntb + 1) * 16 + l15;

  v8f acc0 = v8zero();
  v8f acc1 = v8zero();

  for (int k0 = 0; k0 < K; k0 += 4) {
    v2f a;
    a.x = Arow[k0 + koff];
    a.y = Arow[k0 + koff + 1];
    const float* bp = Bm + (size_t)(k0 + koff) * ldb;
    v2f b0, b1;
    b0.x = bp[bn0];
    b0.y = bp[ldb + bn0];
    b1.x = bp[bn1];
    b1.y = bp[ldb + bn1];
    acc0 = wmma4(a, b0, acc0);
    acc1 = wmma4(a, b1, acc1);
    if ((k0 & 63) == 0 && (k0 + 68) < K) {
      // pull the next B panel toward L0/L2 (global_prefetch_b8)
      __builtin_prefetch(bp + (size_t)64 * ldb + bn0, 0, 1);
      __builtin_prefetch(bp + (size_t)64 * ldb + bn1, 0, 1);
    }
  }

  const int mb = m0 + mt * 16 + (lane >> 4) * 8;
  const float bs0 = (bias1 ? bias1[bn0] : 0.f) + (bias2 ? bias2[bn0] : 0.f);
  const float bs1 = (bias1 ? bias1[bn1] : 0.f) + (bias2 ? bias2[bn1] : 0.f);
#pragma unroll
  for (int i = 0; i < 8; ++i) {
    const size_t m = (size_t)(mb + i);
    C[m * ldc + bn0] = acc0[i] + bs0;
    C[m * ldc + bn1] = acc1[i] + bs1;
  }
}

// ---------------------------------------------------------------------------
// Encoder recurrence: h = tanh(xproj[:,s,:] + h @ Whh), 128 sequential steps.
// One workgroup, 512 threads = 16 wave32s, h kept transposed in 64KB LDS
// (hbuf[k*32+m]) so A-fragment ds loads are near conflict-free.
// Each wave computes a 16x64 strip of the 32x512 h_new (4 WMMA tiles).
// ---------------------------------------------------------------------------
__global__ __launch_bounds__(512) void enc_scan(const float* __restrict__ xproj,
                                                const float* __restrict__ Whh,
                                                float* __restrict__ enc_hs) {
  __shared__ float hbuf[HID * BATCH];  // [k][m], 64KB
  const int tid = threadIdx.x;
  const int lane = tid & 31;
  const int wave = tid >> 5;  // 0..15

  for (int i = tid; i < HID * BATCH; i += 512) hbuf[i] = 0.f;
  __syncthreads();

  const int mt = wave >> 3;  // 0..1
  const int ntq = wave & 7;  // 0..7 -> cols ntq*64 .. +64
  const int l15 = lane & 15;
  const int koff = (lane >> 4) * 2;
  const int am = mt * 16 + l15;
  const int mb = mt * 16 + (lane >> 4) * 8;
  int bn[4];
#pragma unroll
  for (int j = 0; j < 4; ++j) bn[j] = ntq * 64 + j * 16 + l15;

  for (int s = 0; s < SRCLEN; ++s) {
    v8f acc[4];
#pragma unroll
    for (int j = 0; j < 4; ++j) {
      acc[j] = v8zero();
#pragma unroll
      for (int i = 0; i < 8; ++i)
        acc[j][i] = xproj[((size_t)(mb + i) * SRCLEN + s) * HID + bn[j]];
    }
    for (int k0 = 0; k0 < HID; k0 += 4) {
      v2f a;
      a.x = hbuf[(k0 + koff) * BATCH + am];
      a.y = hbuf[(k0 + koff + 1) * BATCH + am];
      const float* bp = Whh + (size_t)(k0 + koff) * HID;
#pragma unroll
      for (int j = 0; j < 4; ++j) {
        v2f b;
        b.x = bp[bn[j]];
        b.y = bp[HID + bn[j]];
        acc[j] = wmma4(a, b, acc[j]);
      }
    }
    __syncthreads();  // everyone finished reading h(s)
#pragma unroll
    for (int j = 0; j < 4; ++j) {
#pragma unroll
      for (int i = 0; i < 8; ++i) {
        const float v = tanhf(acc[j][i]);
        const int m = mb + i;
        hbuf[bn[j] * BATCH + m] = v;  // h(s+1), transposed
        enc_hs[((size_t)m * SRCLEN + s) * HID + bn[j]] = v;
      }
    }
    __syncthreads();  // h(s+1) visible
  }
}

// ---------------------------------------------------------------------------
// Decoder: 64 sequential steps of { proj = h@Ws ; energy/softmax ; ctx ;
// h = tanh(tproj + ctx@Wc + h@Whh) }.  One workgroup, 16 wave32s.
// h / proj / ctx live in global scratch (volatile + threadfence between
// phases); attention weights (32x128) live in 16KB LDS.
// Writes h_new and ctx directly into xcat[(b*T+t), 0:512 / 512:1024] for the
// deferred batched output GEMM.
// ---------------------------------------------------------------------------
__global__ __launch_bounds__(512) void dec_scan(
    const float* __restrict__ enc_hs,    // [B*S, H]
    const float* __restrict__ enc_proj,  // [B*S, H]
    const float* __restrict__ Wattn_s,   // [H, H]
    const float* __restrict__ attn_v,    // [H]
    const float* __restrict__ Wc,        // dec_wih rows 512..1023, [H, H]
    const float* __restrict__ Whh,       // [H, H]
    const float* __restrict__ tproj,     // [B*T, H]
    float* proj_g, float* h_g, float* xcat_g) {
  __shared__ float attn[BATCH * SRCLEN];  // 16KB

  volatile float* hs = h_g;
  volatile float* pj = proj_g;
  volatile float* xc = xcat_g;

  const int tid = threadIdx.x;
  const int lane = tid & 31;
  const int wave = tid >> 5;  // 0..15

  // h0 = enc_hs[:, S-1, :]
  for (int o = tid; o < BATCH * HID; o += 512) {
    const int b = o >> 9, h = o & 511;
    hs[o] = enc_hs[((size_t)(b * SRCLEN + SRCLEN - 1)) * HID + h];
  }
  __threadfence();
  __syncthreads();

  float vr[16];
#pragma unroll
  for (int i = 0; i < 16; ++i) vr[i] = attn_v[lane + 32 * i];

  const int mt = wave >> 3;
  const int ntq = wave & 7;
  const int l15 = lane & 15;
  const int koff = (lane >> 4) * 2;
  const int am = mt * 16 + l15;
  const int mb = mt * 16 + (lane >> 4) * 8;
  int bn[4];
#pragma unroll
  for (int j = 0; j < 4; ++j) bn[j] = ntq * 64 + j * 16 + l15;

  for (int t = 0; t < TGTLEN; ++t) {
    // ---- phase 1: proj = h @ attn_ws ------------------------------------
    {
      v8f acc[4];
#pragma unroll
      for (int j = 0; j < 4; ++j) acc[j] = v8zero();
      for (int k0 = 0; k0 < HID; k0 += 4) {
        v2f a;
        a.x = hs[am * HID + k0 + koff];
        a.y = hs[am * HID + k0 + koff + 1];
        const float* bp = Wattn_s + (size_t)(k0 + koff) * HID;
#pragma unroll
        for (int j = 0; j < 4; ++j) {
          v2f b;
          b.x = bp[bn[j]];
          b.y = bp[HID + bn[j]];
          acc[j] = wmma4(a, b, acc[j]);
        }
      }
#pragma unroll
      for (int j = 0; j < 4; ++j)
#pragma unroll
        for (int i = 0; i < 8; ++i) pj[(mb + i) * HID + bn[j]] = acc[j][i];
    }
    __threadfence();
    __syncthreads();

    // ---- phase 2: energy[b,s] = sum_k tanh(enc_proj+proj) * v -----------
    for (int r = wave; r < BATCH * SRCLEN; r += 16) {
      const int b = r >> 7;  // r = b*128 + s
      const float* ep = enc_proj + (size_t)r * HID;
      float part = 0.f;
#pragma unroll
      for (int i = 0; i < 16; ++i) {
        const int k = lane + 32 * i;
        part += tanhf(ep[k] + pj[b * HID + k]) * vr[i];
      }
#pragma unroll
      for (int off = 16; off; off >>= 1)
        part += __shfl_xor(part, off, 32);
      if (lane == 0) attn[r] = part;
    }
    __syncthreads();

    // ---- phase 3: softmax over s (one wave per row) ---------------------
    for (int b = wave; b < BATCH; b += 16) {
      float v4[4];
#pragma unroll
      for (int i = 0; i < 4; ++i) v4[i] = attn[b * SRCLEN + lane + 32 * i];
      float mx = fmaxf(fmaxf(v4[0], v4[1]), fmaxf(v4[2], v4[3]));
#pragma unroll
      for (int off = 16; off; off >>= 1)
        mx = fmaxf(mx, __shfl_xor(mx, off, 32));
      float sm = 0.f;
#pragma unroll
      for (int i = 0; i < 4; ++i) {
        v4[i] = expf(v4[i] - mx);
        sm += v4[i];
      }
#pragma unroll
      for (int off = 16; off; off >>= 1)
        sm += __shfl_xor(sm, off, 32);
      const float inv = 1.f / sm;
#pragma unroll
      for (int i = 0; i < 4; ++i)
        attn[b * SRCLEN + lane + 32 * i] = v4[i] * inv;
    }
    __syncthreads();

    // ---- phase 4: ctx = attn @ enc_hs  -> xcat[.., 512:1024] ------------
    for (int o = tid; o < BATCH * HID; o += 512) {
      const int b = o >> 9, h = o & 511;
      const float* eh = enc_hs + ((size_t)b * SRCLEN) * HID + h;
      float s = 0.f;
      for (int si = 0; si < SRCLEN; ++si)
        s += attn[b * SRCLEN + si] * eh[(size_t)si * HID];
      xc[((size_t)(b * TGTLEN + t)) * (2 * HID) + HID + h] = s;
    }
    __threadfence();
    __syncthreads();

    // ---- phase 5: h_new = tanh(tproj + ctx@Wc + h@Whh) ------------------
    {
      v8f acc[4];
#pragma unroll
      for (int j = 0; j < 4; ++j) {
        acc[j] = v8zero();
#pragma unroll
        for (int i = 0; i < 8; ++i)
          acc[j][i] = tproj[((size_t)(mb + i) * TGTLEN + t) * HID + bn[j]];
      }
      for (int k0 = 0; k0 < HID; k0 += 4) {  // + h @ Whh
        v2f a;
        a.x = hs[am * HID + k0 + koff];
        a.y = hs[am * HID + k0 + koff + 1];
        const float* bp = Whh + (size_t)(k0 + koff) * HID;
#pragma unroll
        for (int j = 0; j < 4; ++j) {
          v2f b;
          b.x = bp[bn[j]];
          b.y = bp[HID + bn[j]];
          acc[j] = wmma4(a, b, acc[j]);
        }
      }
      const size_t crow = ((size_t)(am * TGTLEN + t)) * (2 * HID) + HID;
      for (int k0 = 0; k0 < HID; k0 += 4) {  // + ctx @ Wc
        v2f a;
        a.x = xc[crow + k0 + koff];
        a.y = xc[crow + k0 + koff + 1];
        const float* bp = Wc + (size_t)(k0 + koff) * HID;
#pragma unroll
        for (int j = 0; j < 4; ++j) {
          v2f b;
          b.x = bp[bn[j]];
          b.y = bp[HID + bn[j]];
          acc[j] = wmma4(a, b, acc[j]);
        }
      }
      __syncthreads();  // all reads of h(t) complete before overwrite
#pragma unroll
      for (int j = 0; j < 4; ++j)
#pragma unroll
        for (int i = 0; i < 8; ++i) {
          const float v = tanhf(acc[j][i]);
          const int m = mb + i;
          xc[((size_t)(m * TGTLEN + t)) * (2 * HID) + bn[j]] = v;
          hs[m * HID + bn[j]] = v;
        }
    }
    __threadfence();
    __syncthreads();
  }
}

// ---------------------------------------------------------------------------
// In-place row-wise log_softmax over 32000 columns; one block per row.
// ---------------------------------------------------------------------------
__global__ __launch_bounds__(256) void logsoftmax_rows(float* __restrict__ out,
                                                       int ncols) {
  __shared__ float red[256];
  const int tid = threadIdx.x;
  float* p = out + (size_t)blockIdx.x * ncols;

  float mx = -3.4e38f;
  for (int i = tid; i < ncols; i += 256) mx = fmaxf(mx, p[i]);
  red[tid] = mx;
  __syncthreads();
  for (int s = 128; s; s >>= 1) {
    if (tid < s) red[tid] = fmaxf(red[tid], red[tid + s]);
    __syncthreads();
  }
  mx = red[0];
  __syncthreads();

  float sm = 0.f;
  for (int i = tid; i < ncols; i += 256) sm += expf(p[i] - mx);
  red[tid] = sm;
  __syncthreads();
  for (int s = 128; s; s >>= 1) {
    if (tid < s) red[tid] += red[tid + s];
    __syncthreads();
  }
  const float l = logf(red[0]);
  __syncthreads();

  for (int i = tid; i < ncols; i += 256) p[i] = p[i] - mx - l;
}

// ---------------------------------------------------------------------------
extern "C" void kernel_launch(void* const* d_in, const int* in_sizes, int n_in,
                              void* d_out, int out_size, void* d_ws,
                              size_t ws_size, hipStream_t stream) {
  (void)in_sizes;
  (void)n_in;
  (void)out_size;
  (void)ws_size;

  const int* src = (const int*)d_in[0];        // [B,S]
  const int* tgt = (const int*)d_in[1];        // [B,T]
  const float* enc_embed = (const float*)d_in[2];
  const float* enc_wih = (const float*)d_in[3];
  const float* enc_bih = (const float*)d_in[4];
  const float* enc_whh = (const float*)d_in[5];
  const float* enc_bhh = (const float*)d_in[6];
  const float* dec_embed = (const float*)d_in[7];
  const float* attn_wh = (const float*)d_in[8];
  const float* attn_ws = (const float*)d_in[9];
  const float* attn_v = (const float*)d_in[10];
  const float* dec_wih = (const float*)d_in[11];  // [1024, 512]
  const float* dec_bih = (const float*)d_in[12];
  const float* dec_whh = (const float*)d_in[13];
  const float* dec_bhh = (const float*)d_in[14];
  const float* out_w = (const float*)d_in[15];  // [1024, 32000]
  const float* out_b = (const float*)d_in[16];
  float* out = (float*)d_out;  // [B,T,V]

  const int RS = BATCH * SRCLEN;  // 4096
  const int RT = BATCH * TGTLEN;  // 2048

  float* ws = (float*)d_ws;
  float* xproj = ws;                       // [RS, H]
  float* enc_hs = xproj + (size_t)RS * HID;
  float* enc_proj = enc_hs + (size_t)RS * HID;
  float* tproj = enc_proj + (size_t)RS * HID;    // [RT, H]
  float* xcat = tproj + (size_t)RT * HID;        // [RT, 2H]
  float* proj_g = xcat + (size_t)RT * 2 * HID;   // [B, H]
  float* h_g = proj_g + (size_t)BATCH * HID;     // [B, H]

  // 1) xproj = enc_embed[src] @ enc_wih + (bih + bhh)
  gemm_tile<<<dim3(RS / 32, HID / 128), 256, 0, stream>>>(
      enc_embed, src, EMB, enc_wih, HID, enc_bih, enc_bhh, xproj, HID, RS, EMB,
      HID);
  // 2) tproj = dec_embed[tgt] @ dec_wih[:512] + (dec_bih + dec_bhh)
  gemm_tile<<<dim3(RT / 32, HID / 128), 256, 0, stream>>>(
      dec_embed, tgt, EMB, dec_wih, HID, dec_bih, dec_bhh, tproj, HID, RT, EMB,
      HID);
  // 3) encoder recurrence -> enc_hs
  enc_scan<<<1, 512, 0, stream>>>(xproj, enc_whh, enc_hs);
  // 4) enc_proj = enc_hs @ attn_wh
  gemm_tile<<<dim3(RS / 32, HID / 128), 256, 0, stream>>>(
      enc_hs, nullptr, HID, attn_wh, HID, nullptr, nullptr, enc_proj, HID, RS,
      HID, HID);
  // 5) decoder recurrence + attention -> xcat = [h_new | ctx]
  dec_scan<<<1, 512, 0, stream>>>(enc_hs, enc_proj, attn_ws, attn_v,
                                  dec_wih + (size_t)HID * HID, dec_whh, tproj,
                                  proj_g, h_g, xcat);
  // 6) logits = xcat @ out_w + out_b  (134 GFLOP, batched over all B*T rows)
  gemm_tile<<<dim3(RT / 32, VOCAB / 128), 256, 0, stream>>>(
      xcat, nullptr, 2 * HID, out_w, VOCAB, out_b, nullptr, out, VOCAB, RT,
      2 * HID, VOCAB);
  // 7) in-place log_softmax
  logsoftmax_rows<<<RT, 256, 0, stream>>>(out, VOCAB);
}